// Codebook_34961033790147
// MI455X (gfx1250) — compile-verified
//
#include <hip/hip_runtime.h>

typedef __attribute__((ext_vector_type(2))) float v2f;
typedef __attribute__((ext_vector_type(8))) float v8f;
typedef __attribute__((ext_vector_type(4))) int   v4i;

#define DIM  256
#define NTOK 32768
#define KCB  8192
#define ETS  260   // padded LDS row stride (words): float4-aligned, conflict-free frag reads

// gfx1250 async global->LDS path (ASYNCcnt-tracked), if the toolchain exposes it.
#if defined(__AMDGCN__) && __has_builtin(__builtin_amdgcn_global_load_async_to_lds_b128)
#define HAVE_ASYNC_LDS 1
#else
#define HAVE_ASYNC_LDS 0
#endif

#if HAVE_ASYNC_LDS
// Builtin prototype (from hipcc diagnostic): (v4i __device__*, ..., int, int)
__device__ __forceinline__ void async_fill_b128(const float* gsrc, float* ldst) {
    __builtin_amdgcn_global_load_async_to_lds_b128(
        (__attribute__((address_space(1))) v4i*)gsrc,
        (__attribute__((address_space(3))) v4i*)ldst,
        0, 0);
}
#endif

// ---------------- e_sq = rowwise ||E||^2 ----------------
__global__ __launch_bounds__(256) void esq_kernel(const float* __restrict__ E,
                                                  float* __restrict__ esq) {
    __shared__ float red[256];
    const int k = blockIdx.x;
    float v = E[k * DIM + threadIdx.x];
    red[threadIdx.x] = v * v;
    __syncthreads();
    for (int s = 128; s > 0; s >>= 1) {
        if (threadIdx.x < s) red[threadIdx.x] += red[threadIdx.x + s];
        __syncthreads();
    }
    if (threadIdx.x == 0) esq[k] = red[0];
}

// ---------------- WMMA distance + argmin ----------------
// Tile: M = 16 codebook rows per chunk, N = 16 tokens per wave, K-loop over D.
// Block = 8 waves = 128 tokens. Grid = NTOK/128. LDS double-buffered embedding tiles.
__global__ __launch_bounds__(256) void argmin_kernel(const float* __restrict__ Z,   // (D, N)
                                                     const float* __restrict__ E,   // (K, D)
                                                     const float* __restrict__ esq, // (K,)
                                                     int* __restrict__ idxOut) {
    __shared__ float eTile[2][16 * ETS];

    const int lane = threadIdx.x & 31;
    const int wave = threadIdx.x >> 5;
    const int tl   = lane & 15;        // token within wave tile / codebook row within A frag
    const int hi   = lane >> 4;        // 0: K=0,1 & M=0..7   1: K=2,3 & M=8..15
    const int n    = blockIdx.x * 128 + wave * 16 + tl;

    // Per-thread staging coordinates (16KB tile, 4x b128 per thread)
    int kr[4], dc[4];
    #pragma unroll
    for (int i = 0; i < 4; ++i) {
        const int f = i * 1024 + threadIdx.x * 4;
        kr[i] = f >> 8;
        dc[i] = f & 255;
    }

    // B operand: z column for token n, resident in 128 VGPRs for the whole K loop.
    v2f b[64];
    #pragma unroll
    for (int j = 0; j < 64; ++j) {
        const int d0 = 4 * j + 2 * hi;
        b[j].x = Z[(size_t)(d0 + 0) * NTOK + n];
        b[j].y = Z[(size_t)(d0 + 1) * NTOK + n];
    }

#if HAVE_ASYNC_LDS
    // Prologue: DMA chunk 0 straight into LDS buffer 0.
    #pragma unroll
    for (int i = 0; i < 4; ++i)
        async_fill_b128(E + kr[i] * DIM + dc[i], &eTile[0][kr[i] * ETS + dc[i]]);
#else
    // Prologue: prefetch chunk 0 into registers.
    float4 stage[4];
    #pragma unroll
    for (int i = 0; i < 4; ++i)
        stage[i] = *(const float4*)(E + kr[i] * DIM + dc[i]);
#endif

    float minVal = 3.402823466e38f;
    int   minIdx = 0;

    int cur = 0;
    for (int k0 = 0; k0 < KCB; k0 += 16, cur ^= 1) {
#if HAVE_ASYNC_LDS
        asm volatile("s_wait_asynccnt 0x0" ::: "memory");  // own async fills of buf[cur] done
        __syncthreads();                                    // everyone's fills done; buf[cur^1] free
        if (k0 + 16 < KCB) {
            const float* Enext = E + (size_t)(k0 + 16) * DIM;
            #pragma unroll
            for (int i = 0; i < 4; ++i)
                async_fill_b128(Enext + kr[i] * DIM + dc[i],
                                &eTile[cur ^ 1][kr[i] * ETS + dc[i]]);
        }
#else
        #pragma unroll
        for (int i = 0; i < 4; ++i)
            *(float4*)(&eTile[cur][kr[i] * ETS + dc[i]]) = stage[i];
        if (k0 + 16 < KCB) {
            const float* Enext = E + (size_t)(k0 + 16) * DIM;
            #pragma unroll
            for (int i = 0; i < 4; ++i)
                stage[i] = *(const float4*)(Enext + kr[i] * DIM + dc[i]);
        }
        __syncthreads();   // buf[cur] visible; lagging readers of buf[cur^1] already arrived
#endif

        const float* tile = eTile[cur];
        const int kbase = k0 + 8 * hi;
        const float4 eq0 = *(const float4*)(esq + kbase);      // issued early, used late
        const float4 eq1 = *(const float4*)(esq + kbase + 4);

        // Two independent WMMA chains + one-stage software pipeline on A fragments.
        v8f acc0 = {0.f, 0.f, 0.f, 0.f, 0.f, 0.f, 0.f, 0.f};
        v8f acc1 = {0.f, 0.f, 0.f, 0.f, 0.f, 0.f, 0.f, 0.f};
        v2f a0, a1, na0, na1;
        a0.x = tile[tl * ETS + 0 + 2 * hi];
        a0.y = tile[tl * ETS + 1 + 2 * hi];
        a1.x = tile[tl * ETS + 4 + 2 * hi];
        a1.y = tile[tl * ETS + 5 + 2 * hi];
        #pragma unroll
        for (int j = 0; j < 64; j += 2) {
            if (j + 2 < 64) {
                const int d2 = 4 * (j + 2) + 2 * hi;
                na0.x = tile[tl * ETS + d2];
                na0.y = tile[tl * ETS + d2 + 1];
                na1.x = tile[tl * ETS + d2 + 4];
                na1.y = tile[tl * ETS + d2 + 5];
            }
            acc0 = __builtin_amdgcn_wmma_f32_16x16x4_f32(
                false, a0, false, b[j],     (short)0, acc0, false, false);
            acc1 = __builtin_amdgcn_wmma_f32_16x16x4_f32(
                false, a1, false, b[j + 1], (short)0, acc1, false, false);
            a0 = na0; a1 = na1;
        }

        // dist = ||e_k||^2 - 2*dot(e_k, z_n)   (||z||^2 constant per token, dropped)
        const float eqv[8] = {eq0.x, eq0.y, eq0.z, eq0.w, eq1.x, eq1.y, eq1.z, eq1.w};
        #pragma unroll
        for (int m = 0; m < 8; ++m) {
            const float dist = eqv[m] - 2.0f * (acc0[m] + acc1[m]);
            if (dist < minVal) { minVal = dist; minIdx = kbase + m; }  // k ascending: < keeps first
        }
    }

    // Merge the two half-wave candidates (M=0..7 vs M=8..15), tie -> lower index.
    const float oVal = __shfl_xor(minVal, 16, 32);
    const int   oIdx = __shfl_xor(minIdx, 16, 32);
    if (oVal < minVal || (oVal == minVal && oIdx < minIdx)) { minVal = oVal; minIdx = oIdx; }
    if (hi == 0) idxOut[n] = minIdx;
}

// ---------------- z_q(D,N) gather: z_q[:, n] = E[idx[n]] ----------------
__global__ __launch_bounds__(256) void gather_kernel(const float* __restrict__ E,
                                                     const int* __restrict__ idx,
                                                     float* __restrict__ zq) {
    __shared__ int sIdx[64];
    const int n0 = blockIdx.x * 64;
    if (threadIdx.x < 64) sIdx[threadIdx.x] = idx[n0 + threadIdx.x];
    __syncthreads();
    const int tx = threadIdx.x & 63;
    const int d0 = threadIdx.x >> 6;           // 0..3
    const float* erow = E + (size_t)sIdx[tx] * DIM;
    #pragma unroll
    for (int i = 0; i < 64; ++i) {
        const int d = d0 + i * 4;
        zq[(size_t)d * NTOK + n0 + tx] = erow[d];
    }
}

extern "C" void kernel_launch(void* const* d_in, const int* in_sizes, int n_in,
                              void* d_out, int out_size, void* d_ws, size_t ws_size,
                              hipStream_t stream) {
    const float* z_e = (const float*)d_in[0];   // (D, N) fp32
    const float* emb = (const float*)d_in[1];   // (K, D) fp32
    float* zq     = (float*)d_out;                                  // first D*N floats
    int*   idxOut = (int*)((float*)d_out + (size_t)DIM * NTOK);     // then N int32
    float* esq    = (float*)d_ws;                                   // K floats scratch

    esq_kernel   <<<KCB,        256, 0, stream>>>(emb, esq);
    argmin_kernel<<<NTOK / 128, 256, 0, stream>>>(z_e, emb, esq, idxOut);
    gather_kernel<<<NTOK / 64,  256, 0, stream>>>(emb, idxOut, zq);
}